// InterEinsumTC_17600775979373
// MI455X (gfx1250) — compile-verified
//
#include <hip/hip_runtime.h>
#include <stdint.h>

// comb einsum: out[b,c,i,l] = (1/3) * sum_{m=0..8} T1[b,c,i,m] * T2[b,c,m,l]
// (m = j*3+k).  One pair per thread; each block stages its contiguous tiles
// through LDS with CDNA5 async global<->LDS DMA (ASYNCcnt path) so all HBM
// traffic is coalesced 128-bit, non-temporal (stream-once) accesses.
//
// Roofline: 0.64 FLOP/byte -> purely HBM-bound (~23 us at 23.3 TB/s for the
// reference shapes).  WMMA (16x16xK) is deliberately not used: block-diagonal
// packing of 3x9x3 pair-GEMMs into V_WMMA_F32_16X16X4_F32 costs more wave
// instructions than the VALU FMA path at 13% MACC utilization and cannot
// affect the memory-bound runtime.

#define PPB 256                                // pairs per block
#define ELEMS 27                               // floats per tensor per pair
#define REGION_CHUNKS ((PPB * ELEMS * 4) / 16) // 1728 x 16B per input
#define TOTAL_CHUNKS (2 * REGION_CHUNKS)       // 3456
#define OUT_CHUNKS ((PPB * 9 * 4) / 16)        // 576 x 16B output

__global__ __launch_bounds__(PPB) void einsum33_async_kernel(
    const float* __restrict__ T1, const float* __restrict__ T2,
    float* __restrict__ out)
{
    __shared__ __align__(16) float sAB[2 * PPB * ELEMS]; // 55296 bytes
    const int tid = threadIdx.x;
    const size_t pair0 = (size_t)blockIdx.x * PPB;

    const uint64_t gA = (uint64_t)(uintptr_t)(T1 + pair0 * ELEMS);
    const uint64_t gB = (uint64_t)(uintptr_t)(T2 + pair0 * ELEMS);
    const uint32_t ldsBase = (uint32_t)(uintptr_t)&sAB[0];

    // ---- async cooperative copy: global -> LDS, 16B per lane per issue ----
    // Stream-once data: non-temporal hint keeps it from occupying L2 at RT.
    #pragma unroll
    for (int it = 0; it < (TOTAL_CHUNKS + PPB - 1) / PPB; ++it) {
        int c = tid + it * PPB;
        if (c < TOTAL_CHUNKS) {
            const bool isB = (c >= REGION_CHUNKS);
            const uint64_t base = isB ? gB : gA;
            const uint32_t local = isB ? (uint32_t)(c - REGION_CHUNKS)
                                       : (uint32_t)c;
            const uint64_t gaddr = base + (uint64_t)local * 16u;
            const uint32_t laddr = ldsBase + (uint32_t)c * 16u;
            asm volatile("global_load_async_to_lds_b128 %0, %1, off th:TH_LOAD_NT"
                         :: "v"(laddr), "v"(gaddr)
                         : "memory");
        }
    }
    asm volatile("s_wait_asynccnt 0" ::: "memory");
    __syncthreads();

    // ---- per-pair 3x9 * 9x3 contraction from LDS (bank-conflict free:
    //      gcd(27,64)=1 so 27*t dword addressing is conflict-free) ----
    const float* sA = sAB + tid * ELEMS;
    const float* sB = sAB + PPB * ELEMS + tid * ELEMS;

    float a[ELEMS], b[ELEMS];
    #pragma unroll
    for (int i = 0; i < ELEMS; ++i) a[i] = sA[i];
    #pragma unroll
    for (int i = 0; i < ELEMS; ++i) b[i] = sB[i];

    float acc[9];
    #pragma unroll
    for (int i = 0; i < 9; ++i) acc[i] = 0.0f;

    #pragma unroll
    for (int m = 0; m < 9; ++m) {
        #pragma unroll
        for (int i = 0; i < 3; ++i) {
            const float av = a[i * 9 + m];
            #pragma unroll
            for (int l = 0; l < 3; ++l) {
                acc[i * 3 + l] = fmaf(av, b[m * 3 + l], acc[i * 3 + l]);
            }
        }
    }

    const float NORM = 1.0f / 3.0f;  // 1/sqrt(3^2)

    // ---- stage results in LDS (reuse A region), then async LDS -> global ----
    __syncthreads();  // everyone done reading sAB before we overwrite it
    #pragma unroll
    for (int q = 0; q < 9; ++q) sAB[tid * 9 + q] = acc[q] * NORM;
    __syncthreads();  // all results visible in LDS

    const uint64_t gOut = (uint64_t)(uintptr_t)(out + pair0 * 9);
    #pragma unroll
    for (int it = 0; it < (OUT_CHUNKS + PPB - 1) / PPB; ++it) {
        int c = tid + it * PPB;
        if (c < OUT_CHUNKS) {
            const uint64_t gaddr = gOut + (uint64_t)c * 16u;
            const uint32_t laddr = ldsBase + (uint32_t)c * 16u;
            asm volatile("global_store_async_from_lds_b128 %0, %1, off th:TH_STORE_NT"
                         :: "v"(gaddr), "v"(laddr)
                         : "memory");
        }
    }
    // Drain the LDS->global DMA before the workgroup's LDS can be retired.
    asm volatile("s_wait_asynccnt 0" ::: "memory");
}

// Scalar tail for n_pairs % PPB (zero blocks for the reference shapes).
__global__ void einsum33_tail_kernel(const float* __restrict__ T1,
                                     const float* __restrict__ T2,
                                     float* __restrict__ out,
                                     long first_pair, long n_pairs)
{
    long p = first_pair + blockIdx.x * blockDim.x + threadIdx.x;
    if (p >= n_pairs) return;
    const float* a = T1 + p * ELEMS;
    const float* b = T2 + p * ELEMS;
    float* o = out + p * 9;
    #pragma unroll
    for (int i = 0; i < 3; ++i) {
        #pragma unroll
        for (int l = 0; l < 3; ++l) {
            float acc = 0.0f;
            #pragma unroll
            for (int m = 0; m < 9; ++m)
                acc = fmaf(a[i * 9 + m], b[m * 3 + l], acc);
            o[i * 3 + l] = acc * (1.0f / 3.0f);
        }
    }
}

extern "C" void kernel_launch(void* const* d_in, const int* in_sizes, int n_in,
                              void* d_out, int out_size, void* d_ws, size_t ws_size,
                              hipStream_t stream) {
    const float* T1 = (const float*)d_in[0];
    const float* T2 = (const float*)d_in[1];
    float* out = (float*)d_out;

    const long n_pairs = (long)in_sizes[0] / ELEMS;   // B*C = 2^21 for ref shapes
    const long full_blocks = n_pairs / PPB;           // 8192 for ref shapes
    const long rem = n_pairs - full_blocks * PPB;     // 0 for ref shapes

    if (full_blocks > 0) {
        einsum33_async_kernel<<<(int)full_blocks, PPB, 0, stream>>>(T1, T2, out);
    }
    if (rem > 0) {
        const int tb = 256;
        einsum33_tail_kernel<<<(int)((rem + tb - 1) / tb), tb, 0, stream>>>(
            T1, T2, out, full_blocks * PPB, n_pairs);
    }
}